// ClassifierModel_87883620811309
// MI455X (gfx1250) — compile-verified
//
#include <hip/hip_runtime.h>
#include <hip/hip_bf16.h>

#define N_PROP 16384
#define MAXL   128
#define BATCH  32

typedef __attribute__((ext_vector_type(2))) float v2f;
typedef __attribute__((ext_vector_type(8))) float v8f;

// ---------------------------------------------------------------------------
// Kernel 0: zero the double accumulator in workspace
// ---------------------------------------------------------------------------
__global__ void init_acc_kernel(double* acc) { *acc = 0.0; }

// ---------------------------------------------------------------------------
// Kernel 1: per-(image,label) argmax of IoU over all 16384 ROIs.
// grid = (MAXL, BATCH), block = 256. Non-contracted f32 math so equal-IoU
// ties (esp. all-zero rows) resolve to the first index like jnp.argmax.
// ---------------------------------------------------------------------------
__global__ void argmax_iou_kernel(const float* __restrict__ roi,
                                  const float* __restrict__ labels,
                                  int* __restrict__ match) {
  const int l   = blockIdx.x;
  const int b   = blockIdx.y;
  const int tid = threadIdx.x;

  const float4 lb = ((const float4*)(labels + ((size_t)b * MAXL + l) * 4))[0];
  const float ax1 = lb.x, ay1 = lb.y, aw = lb.z, ah = lb.w;
  const float ax2 = __fadd_rn(ax1, aw);
  const float ay2 = __fadd_rn(ay1, ah);
  const float aArea = __fmul_rn(aw, ah);

  const float4* roib = (const float4*)(roi + (size_t)b * N_PROP * 4);

  float best = -1.0f;
  int   bidx = 0;
  for (int n = tid; n < N_PROP; n += blockDim.x) {
    const float4 r = roib[n];
    const float rx = __fmul_rn(r.x, 16.0f);
    const float ry = __fmul_rn(r.y, 16.0f);
    const float rw = __fmul_rn(r.z, 16.0f);
    const float rh = __fmul_rn(r.w, 16.0f);
    const float bx2 = __fadd_rn(rx, rw);
    const float by2 = __fadd_rn(ry, rh);
    const float iw = fmaxf(__fsub_rn(fminf(ax2, bx2), fmaxf(ax1, rx)), 0.0f);
    const float ih = fmaxf(__fsub_rn(fminf(ay2, by2), fmaxf(ay1, ry)), 0.0f);
    const float inter = __fmul_rn(iw, ih);
    const float uni   = __fsub_rn(__fadd_rn(aArea, __fmul_rn(rw, rh)), inter);
    const float iou   = inter / fmaxf(uni, 1e-12f);
    if (iou > best) { best = iou; bidx = n; }  // n ascending -> keeps first max
  }

  __shared__ float sV[256];
  __shared__ int   sI[256];
  sV[tid] = best; sI[tid] = bidx;
  __syncthreads();
  for (int s = blockDim.x >> 1; s > 0; s >>= 1) {
    if (tid < s) {
      const float ov = sV[tid + s];
      const int   oi = sI[tid + s];
      if (ov > sV[tid] || (ov == sV[tid] && oi < sI[tid])) { sV[tid] = ov; sI[tid] = oi; }
    }
    __syncthreads();
  }
  if (tid == 0) match[b * MAXL + l] = sI[0];
}

// ---------------------------------------------------------------------------
// Kernel 2: sum of squares via V_WMMA_F32_16X16X4_F32 (f32-exact).
// A = all-ones 16x4, B = 64 squared elements per wave iteration, C accumulates.
// All D rows are identical (= column sums), so c[0] across the 32 lanes holds
// two copies of one row: total = waveReduce(c[0]) * 0.5.
// ---------------------------------------------------------------------------
__global__ void ssq_wmma_kernel(const float* __restrict__ x, int count64,
                                double coeff, double* __restrict__ acc) {
  const int tid  = threadIdx.x;
  const int lane = tid & 31;
  const int wave = tid >> 5;
  const int wavesPerBlock = blockDim.x >> 5;
  const int gwave  = blockIdx.x * wavesPerBlock + wave;
  const int nwaves = gridDim.x * wavesPerBlock;

  v2f ones; ones[0] = 1.0f; ones[1] = 1.0f;
  v8f c = {0.f, 0.f, 0.f, 0.f, 0.f, 0.f, 0.f, 0.f};

  for (int i = gwave; i < count64; i += nwaves) {
    const float2 v = ((const float2*)(x + (size_t)i * 64))[lane];
    v2f bsq; bsq[0] = v.x * v.x; bsq[1] = v.y * v.y;
    c = __builtin_amdgcn_wmma_f32_16x16x4_f32(
        /*neg_a=*/false, ones, /*neg_b=*/false, bsq,
        /*c_mod=*/(short)0, c, /*reuse_a=*/false, /*reuse_b=*/false);
  }

  float rowv = c[0];
  for (int off = 16; off > 0; off >>= 1) rowv += __shfl_xor(rowv, off, 32);
  if (lane == 0) atomicAdd(acc, (double)(rowv * 0.5f) * coeff);
}

// ---------------------------------------------------------------------------
// Kernel 3: per-image main loss: first-match inversion (128-entry LDS scan),
// Huber on positives, softmax CCE on all proposals. Double block reduction,
// one global f64 atomic add per block.
// ---------------------------------------------------------------------------
__global__ void loss_main_kernel(const float* __restrict__ cls,
                                 const float* __restrict__ bbox,
                                 const float* __restrict__ roi,
                                 const float* __restrict__ labels,
                                 const int* __restrict__ match,
                                 double* __restrict__ acc) {
  const int b   = blockIdx.x;
  const int tid = threadIdx.x;

  __shared__ int    sMatch[MAXL];
  __shared__ int    sValid[MAXL];
  __shared__ float4 sLab[MAXL];
  for (int l = tid; l < MAXL; l += blockDim.x) {
    const float4 lb = ((const float4*)(labels + ((size_t)b * MAXL + l) * 4))[0];
    sLab[l]   = lb;
    sValid[l] = (lb.x != 0.f || lb.y != 0.f || lb.z != 0.f || lb.w != 0.f) ? 1 : 0;
    sMatch[l] = match[b * MAXL + l];
  }
  __syncthreads();

  const float*  clsb = cls  + (size_t)b * 2 * N_PROP;
  const float*  bbb  = bbox + (size_t)b * 4 * N_PROP;
  const float4* roib = (const float4*)(roi + (size_t)b * N_PROP * 4);

  const float log_lo = -16.118095651f;       // log(1e-7)
  const float log_hi = -1.0000000495e-07f;   // log(1 - 1e-7)

  double local = 0.0;
  for (int n = tid; n < N_PROP; n += blockDim.x) {
    // first[n] = min{l : valid[l] && match[l]==n}, else MAXL
    int fl = MAXL;
    for (int l = 0; l < MAXL; ++l) {
      if (sValid[l] && sMatch[l] == n) { fl = l; break; }
    }
    const bool positive = (fl < MAXL);

    // softmax CCE (stable, max-subtracted like jax.nn.softmax)
    const float c0 = clsb[n];
    const float c1 = clsb[N_PROP + n];
    const float m  = fmaxf(c0, c1);
    const float e0 = expf(c0 - m);
    const float e1 = expf(c1 - m);
    const float inv = 1.0f / (e0 + e1);
    const float p0 = e0 * inv;
    const float p1 = e1 * inv;
    float contrib = positive ? -(p0 * log_lo + p1 * log_hi)
                             : -(p0 * log_hi + p1 * log_lo);

    if (positive) {
      const float4 tr = sLab[fl];
      const float4 r  = roib[n];
      const float rx = r.x * 16.0f, ry = r.y * 16.0f;
      const float rw = r.z * 16.0f, rh = r.w * 16.0f;
      const float t0 = (tr.x - rx) / rw;
      const float t1 = (tr.y - ry) / rh;
      const float t2 = logf(fmaxf(tr.z / rw, 1e-10f));
      const float t3 = logf(fmaxf(tr.w / rh, 1e-10f));
      const float b0 = bbb[n];
      const float b1 = bbb[N_PROP + n];
      const float b2 = bbb[2 * N_PROP + n];
      const float b3 = bbb[3 * N_PROP + n];
      float h = 0.0f, e;
      e = t0 - b0; h += (fabsf(e) <= 1.0f) ? 0.5f * e * e : fabsf(e) - 0.5f;
      e = t1 - b1; h += (fabsf(e) <= 1.0f) ? 0.5f * e * e : fabsf(e) - 0.5f;
      e = t2 - b2; h += (fabsf(e) <= 1.0f) ? 0.5f * e * e : fabsf(e) - 0.5f;
      e = t3 - b3; h += (fabsf(e) <= 1.0f) ? 0.5f * e * e : fabsf(e) - 0.5f;
      contrib += h * 0.25f;  // mean over 4 components
    }
    local += (double)contrib;
  }

  __shared__ double sAcc[256];
  sAcc[tid] = local;
  __syncthreads();
  for (int s = blockDim.x >> 1; s > 0; s >>= 1) {
    if (tid < s) sAcc[tid] += sAcc[tid + s];
    __syncthreads();
  }
  if (tid == 0) atomicAdd(acc, sAcc[0]);
}

// ---------------------------------------------------------------------------
// Kernel 4: write the f32 scalar result
// ---------------------------------------------------------------------------
__global__ void finalize_kernel(const double* acc, float* out) {
  *out = (float)(*acc);
}

extern "C" void kernel_launch(void* const* d_in, const int* in_sizes, int n_in,
                              void* d_out, int out_size, void* d_ws, size_t ws_size,
                              hipStream_t stream) {
  (void)in_sizes; (void)n_in; (void)out_size; (void)ws_size;
  const float* cls    = (const float*)d_in[0];   // (32, 32768)
  const float* bbox   = (const float*)d_in[1];   // (32, 65536)
  const float* roi    = (const float*)d_in[2];   // (32, 16384, 4)
  const float* labels = (const float*)d_in[3];   // (32, 128, 4)

  double* acc   = (double*)d_ws;                      // 8B accumulator
  int*    match = (int*)((char*)d_ws + 64);           // 32*128 ints = 16 KB
  float*  out   = (float*)d_out;

  init_acc_kernel<<<1, 1, 0, stream>>>(acc);

  argmax_iou_kernel<<<dim3(MAXL, BATCH), 256, 0, stream>>>(roi, labels, match);

  // 0.5*sum(cls^2)/(10*2N) and 0.5*sum(bbox^2)/(4N), summed over images
  ssq_wmma_kernel<<<64, 256, 0, stream>>>(cls,  BATCH * 2 * N_PROP / 64,
                                          0.5 / (20.0 * (double)N_PROP), acc);
  ssq_wmma_kernel<<<64, 256, 0, stream>>>(bbox, BATCH * 4 * N_PROP / 64,
                                          0.5 / (4.0 * (double)N_PROP), acc);

  loss_main_kernel<<<BATCH, 256, 0, stream>>>(cls, bbox, roi, labels, match, acc);

  finalize_kernel<<<1, 1, 0, stream>>>(acc, out);
}